// EMAAttention_67929202753864
// MI455X (gfx1250) — compile-verified
//
#include <hip/hip_runtime.h>

typedef float v2f __attribute__((ext_vector_type(2)));
typedef float v8f __attribute__((ext_vector_type(8)));

#define BATCH   16
#define CHAN    256
#define LLEN    6400
#define CHUNK   256
#define NCHUNK  (LLEN / CHUNK)   // 25
#define NSEG    5                // segments per row (25 = 5 * 5, exact)
#define SEGCH   (NCHUNK / NSEG)  // 5 chunks per segment

// 0.9^d for 0<=d<16 via square-and-multiply on the 4 bits of d (no indexed
// register array -> no cndmask chains).
__device__ __forceinline__ float pow09_4bit(int d) {
    float p = 1.0f;
    p *= (d & 1) ? 0.9f        : 1.0f;
    p *= (d & 2) ? 0.81f       : 1.0f;
    p *= (d & 4) ? 0.6561f     : 1.0f;
    p *= (d & 8) ? 0.43046721f : 1.0f;
    return p;
}

// One wave32 per (row, segment). Chunk = 256 elements = 16 tiles of 16.
// Intra-tile EMA scan as Y = T*X via v_wmma_f32_16x16x4_f32 (T = 16x16
// lower-triangular decay matrix); cross-tile carries via log-step shuffle
// scan with literal factors. Segments > 0 reconstruct their incoming carry
// from one warm-up chunk (neglected tail decays by 0.9^256 ~ 2e-12).
__global__ __launch_bounds__(32)
void ema_conv_kernel(const float* __restrict__ x,
                     const float* __restrict__ w,
                     const float* __restrict__ bias,
                     float* __restrict__ out) {
    __shared__ float lds[CHUNK + 2];   // [0..255]=chunk, [256]=left halo, [257]=right halo

    const int  lane = threadIdx.x;           // 0..31
    const int  seg  = blockIdx.x;            // 0..NSEG-1
    const int  row  = blockIdx.y;            // b*C + c
    const int  ch   = row & (CHAN - 1);
    const long base = (long)row * LLEN;
    const int  g0   = seg * SEGCH;           // first chunk of this segment

    const float w0 = w[ch * 3 + 0];
    const float w1 = w[ch * 3 + 1] + 1.0f;   // fold "out += x" into center tap
    const float w2 = w[ch * 3 + 2];
    const float bb = bias[ch];

    // A operand: T[i][k] = (k<=i) ? 0.1*0.9^(i-k) : 0, as 4 K-slices of 16x4.
    // 32-bit A 16x4 layout: lanes 0-15 -> K=0 (v0), K=1 (v1); lanes 16-31 -> K=2,3.
    const int mi = lane & 15;                // M for A, N (tile id) for B/C/D
    const int kb = (lane < 16) ? 0 : 2;
    v2f afrag[4];
    #pragma unroll
    for (int s = 0; s < 4; ++s) {
        int k0 = 4 * s + kb;
        afrag[s][0] = (k0     <= mi) ? 0.1f * pow09_4bit((mi - k0)     & 15) : 0.0f;
        afrag[s][1] = (k0 + 1 <= mi) ? 0.1f * pow09_4bit((mi - k0 - 1) & 15) : 0.0f;
    }

    const int   half  = (lane >= 16) ? 8 : 0;
    const int   basep = mi * 16 + half;          // 8 contiguous outputs per lane
    const float hs    = (lane >= 16) ? 0.43046721f : 1.0f;   // 0.9^8 half-offset

    // Stage one chunk (+halos) into LDS (coalesced 2x b128 per lane).
    auto stage = [&](int gc) {
        const long cb = base + (long)gc * CHUNK;
        const float4 va = *(const float4*)(x + cb + lane * 8);
        const float4 vb = *(const float4*)(x + cb + lane * 8 + 4);
        *(float4*)(&lds[lane * 8])     = va;
        *(float4*)(&lds[lane * 8 + 4]) = vb;
        if (lane == 0) lds[CHUNK]     = (gc > 0)          ? x[cb - 1]     : 0.0f;
        if (lane == 1) lds[CHUNK + 1] = (gc + 1 < NCHUNK) ? x[cb + CHUNK] : 0.0f;
        __syncthreads();
    };

    // 16x16x16 f32 matmul: acc = T * X, X[k][t] = lds[t*16 + k].
    auto tile_wmma = [&]() -> v8f {
        v8f acc = {};
        #pragma unroll
        for (int s = 0; s < 4; ++s) {
            int    k0 = 4 * s + kb;
            float2 bp = *(const float2*)(&lds[mi * 16 + k0]);   // 8B-aligned b64
            v2f    bfrag;  bfrag[0] = bp.x;  bfrag[1] = bp.y;
            acc = __builtin_amdgcn_wmma_f32_16x16x4_f32(
                      false, afrag[s], false, bfrag, (short)0, acc, false, false);
        }
        return acc;
    };

    // Inclusive cross-tile scan g[t] = 0.9^16*g[t-1] + s_t (s_t = acc row 15),
    // optionally folding an incoming chunk carry. All factors are literals.
    auto tile_scan = [&](const v8f& acc, float fold_carry) -> float {
        float g = __shfl(acc[7], 16 + mi, 32);   // s_t from M=15 row, lane 16+t
        if (mi == 0) g += 0.1853020188851841f * fold_carry;
        float o;
        o = __shfl(g, (lane - 1) & 31, 32); if (mi >= 1) g = fmaf(0.1853020188851841f,     o, g);
        o = __shfl(g, (lane - 2) & 31, 32); if (mi >= 2) g = fmaf(0.03433683820292512f,    o, g);
        o = __shfl(g, (lane - 4) & 31, 32); if (mi >= 4) g = fmaf(0.0011790184577738583f,  o, g);
        o = __shfl(g, (lane - 8) & 31, 32); if (mi >= 8) g = fmaf(1.3900845237714473e-06f, o, g);
        return g;
    };

    // ---- Incoming segment carry ----
    float carry;
    if (seg == 0) {
        carry = x[base];                     // e_{-1} = x[0]  =>  ema[0] = x[0]
    } else {
        __builtin_prefetch(x + base + (long)g0 * CHUNK + lane * 8, 0, 1);
        stage(g0 - 1);                       // warm-up chunk: scan only, no output
        v8f accw = tile_wmma();
        float gw = tile_scan(accw, 0.0f);
        carry = __shfl(gw, 15, 32);
        __syncthreads();
    }

    // ---- Main segment: SEGCH chunks ----
    for (int c = 0; c < SEGCH; ++c) {
        const int  gc = g0 + c;
        const long cb = base + (long)gc * CHUNK;

        if (gc + 1 < NCHUNK)
            __builtin_prefetch(x + cb + CHUNK + lane * 8, 0, 1);  // global_prefetch_b8

        stage(gc);
        v8f acc = tile_wmma();
        // acc[r] = local scan at position m = r + half within tile t = mi.

        float g = tile_scan(acc, carry);
        const float cin_tile = (mi == 0) ? carry : __shfl(g, (lane - 1) & 31, 32);
        carry = __shfl(g, 15, 32);           // carry out of tile 15 -> next chunk

        // Conv inputs: 10 elements basep-1 .. basep+8 (2x b128 + 2 guarded scalars).
        float xs[10];
        const float4 c0 = *(const float4*)(&lds[basep]);
        const float4 c1 = *(const float4*)(&lds[basep + 4]);
        xs[1] = c0.x; xs[2] = c0.y; xs[3] = c0.z; xs[4] = c0.w;
        xs[5] = c1.x; xs[6] = c1.y; xs[7] = c1.z; xs[8] = c1.w;
        xs[0] = lds[(basep == 0)         ? CHUNK     : (basep - 1)];
        xs[9] = lds[(basep + 8 == CHUNK) ? CHUNK + 1 : (basep + 8)];

        // Outputs: conv + ema (+x folded into w1); EMA coeffs are literals.
        const float cinh = hs * cin_tile;    // 0.9^half * incoming tile carry
        float4 o0, o1;
        #pragma unroll
        for (int r = 0; r < 8; ++r) {
            const float dc = (r == 0) ? 0.9f
                           : (r == 1) ? 0.81f
                           : (r == 2) ? 0.729f
                           : (r == 3) ? 0.6561f
                           : (r == 4) ? 0.59049f
                           : (r == 5) ? 0.531441f
                           : (r == 6) ? 0.4782969f
                           :            0.43046721f;
            float e  = fmaf(dc, cinh, acc[r]);
            float ov = fmaf(w0, xs[r], fmaf(w1, xs[r + 1], fmaf(w2, xs[r + 2], bb + e)));
            if (r < 4) ((float*)&o0)[r] = ov; else ((float*)&o1)[r - 4] = ov;
        }
        *(float4*)(out + cb + basep)     = o0;
        *(float4*)(out + cb + basep + 4) = o1;
        __syncthreads();                     // protect LDS before next-chunk overwrite
    }
}

extern "C" void kernel_launch(void* const* d_in, const int* in_sizes, int n_in,
                              void* d_out, int out_size, void* d_ws, size_t ws_size,
                              hipStream_t stream) {
    const float* x    = (const float*)d_in[0];
    const float* w    = (const float*)d_in[1];
    const float* b    = (const float*)d_in[2];
    float*       outp = (float*)d_out;

    dim3 grid(NSEG, BATCH * CHAN);   // 5 segments x 4096 rows = 20480 wave32's
    dim3 block(32);
    hipLaunchKernelGGL(ema_conv_kernel, grid, block, 0, stream, x, w, b, outp);
}